// FilterbankCell1_74569222193668
// MI455X (gfx1250) — compile-verified
//
#include <hip/hip_runtime.h>
#include <math.h>

typedef __attribute__((ext_vector_type(2))) float v2f;
typedef __attribute__((ext_vector_type(8))) float v8f;

#define TAPS 32
#define ROWS_PER_GROUP 128   // 8 waves/block x 16 rows/wave
#define APAD 36              // LDS pitch: 16B aligned (36*4=144) + bank-conflict-free

// gfx1250 async memory->LDS path (ASYNCcnt tracked); silent fallback if absent.
#if __has_builtin(__builtin_amdgcn_global_load_async_to_lds_b128)
#define USE_ASYNC_LDS 1
// param 1 per clang diagnostic: 'int __attribute__((__vector_size__(16))) __device__ *'
typedef int vi4 __attribute__((__vector_size__(4 * sizeof(int))));
typedef __attribute__((address_space(1))) vi4* gvi4p;   // global (AS1) src
typedef __attribute__((address_space(3))) vi4* lvi4p;   // LDS (AS3) dst
#if __has_builtin(__builtin_amdgcn_s_wait_asynccnt)
#define WAIT_ASYNC() __builtin_amdgcn_s_wait_asynccnt(0)
#else
#define WAIT_ASYNC() asm volatile("s_wait_asynccnt 0x0" ::: "memory")
#endif
#else
#define USE_ASYNC_LDS 0
#define WAIT_ASYNC() ((void)0)
#endif

// out[B,32] = [prev | x][B,33] @ Bmat[33,32]   via V_WMMA_F32_16X16X4_F32
__global__ __launch_bounds__(256)
void laguerre_wmma_kernel(const float* __restrict__ x,
                          const float* __restrict__ prev,
                          const float* __restrict__ relax,
                          float* __restrict__ out,
                          int Btotal, int groupsPerBlock, int numBlocks)
{
    __shared__ float Bsh[32][40];                 // [N][k] coefficient columns
    __shared__ float As[ROWS_PER_GROUP][APAD];    // staged prev rows
    __shared__ float Xs[ROWS_PER_GROUP];          // staged x

    const int tid  = threadIdx.x;
    const int lane = tid & 31;          // wave32 on gfx1250
    const int wave = tid >> 5;
    const int n    = lane & 15;         // N column within 16-wide tile / M row
    const int h    = lane >> 4;         // lane half selects K sub-pair

    const float r    = relax[0];
    const float sr   = sqrtf(r);
    const float s1mr = sqrtf(1.0f - r);

    // One thread per output tap N builds coefficient column Bmat[:,N].
    if (tid < 32) {
        const int N = tid;
        #pragma unroll
        for (int k = 0; k < 40; ++k) Bsh[N][k] = 0.0f;
        Bsh[N][N] = sr;
        float p = sr * sr - 1.0f;
        for (int k = N - 1; k >= 0; --k) { Bsh[N][k] = p; p *= sr; }
        float q = s1mr;
        for (int i = 0; i < N; ++i) q *= sr;
        Bsh[N][32] = q;                  // x-column: s1mr * sr^N
    }
    __syncthreads();

    // Hoist per-lane B fragments (9 K-chunks x 2 N-tiles), reused for all groups.
    v2f bf[9][2];
    #pragma unroll
    for (int c = 0; c < 9; ++c) {
        #pragma unroll
        for (int t = 0; t < 2; ++t) {
            const int N = t * 16 + n;
            const int k = 4 * c + 2 * h;
            v2f tmp; tmp.x = Bsh[N][k]; tmp.y = Bsh[N][k + 1];
            bf[c][t] = tmp;
        }
    }

    for (int g = 0; g < groupsPerBlock; ++g) {
        const int group   = blockIdx.x + g * numBlocks;
        const int rowBase = group * ROWS_PER_GROUP;
        if (rowBase >= Btotal) break;
        const bool fullGroup = (rowBase + ROWS_PER_GROUP) <= Btotal;

        // ---- stage 128 rows of prev + x into LDS ----
        const float4* gp = (const float4*)(prev + (size_t)rowBase * TAPS);
        if (fullGroup) {
#if USE_ASYNC_LDS
            // async DMA: 4x16B per lane, perfectly coalesced, no VGPR staging
            #pragma unroll
            for (int i = 0; i < 4; ++i) {
                const int q   = i * 256 + tid;    // 0..1023 float4s
                const int row = q >> 3, chnk = q & 7;
                __builtin_amdgcn_global_load_async_to_lds_b128(
                    (gvi4p)(gp + q), (lvi4p)&As[row][chnk * 4], 0, 0);
            }
#else
            #pragma unroll
            for (int i = 0; i < 4; ++i) {
                const int q   = i * 256 + tid;
                const int row = q >> 3, chnk = q & 7;
                float4 v = gp[q];
                As[row][chnk * 4 + 0] = v.x;
                As[row][chnk * 4 + 1] = v.y;
                As[row][chnk * 4 + 2] = v.z;
                As[row][chnk * 4 + 3] = v.w;
            }
#endif
            if (tid < ROWS_PER_GROUP) Xs[tid] = x[rowBase + tid];
        } else {
            #pragma unroll
            for (int i = 0; i < 4; ++i) {
                const int q   = i * 256 + tid;
                const int row = q >> 3, chnk = q & 7;
                float4 v = {0.f, 0.f, 0.f, 0.f};
                if (rowBase + row < Btotal) v = gp[q];
                As[row][chnk * 4 + 0] = v.x;
                As[row][chnk * 4 + 1] = v.y;
                As[row][chnk * 4 + 2] = v.z;
                As[row][chnk * 4 + 3] = v.w;
            }
            if (tid < ROWS_PER_GROUP)
                Xs[tid] = (rowBase + tid < Btotal) ? x[rowBase + tid] : 0.0f;
        }

        // prefetch next group's rows while we compute this one
        const long long nrb = (long long)(group + numBlocks) * ROWS_PER_GROUP;
        if (nrb < Btotal)
            __builtin_prefetch(prev + nrb * TAPS + (long long)tid * 16, 0, 0);

        WAIT_ASYNC();          // drain this wave's async->LDS transfers
        __syncthreads();       // all waves' staging visible

        // ---- per-wave 16x32 output tile: 9 K-steps x 2 N-tiles of WMMA f32 ----
        const int grow = wave * 16;
        v8f acc0 = {0.f,0.f,0.f,0.f,0.f,0.f,0.f,0.f};
        v8f acc1 = {0.f,0.f,0.f,0.f,0.f,0.f,0.f,0.f};

        #pragma unroll
        for (int c = 0; c < 9; ++c) {
            v2f a;
            if (c < 8) {
                // A 16x4 layout: lanes 0-15 hold K=4c+0,1; lanes 16-31 K=4c+2,3
                a.x = As[grow + n][4 * c + 2 * h];
                a.y = As[grow + n][4 * c + 2 * h + 1];
            } else {
                // K chunk 8 = {x, 0, 0, 0}
                a.x = (h == 0) ? Xs[grow + n] : 0.0f;
                a.y = 0.0f;
            }
            acc0 = __builtin_amdgcn_wmma_f32_16x16x4_f32(
                false, a, false, bf[c][0], (short)0, acc0, false, false);
            acc1 = __builtin_amdgcn_wmma_f32_16x16x4_f32(
                false, a, false, bf[c][1], (short)0, acc1, false, false);
        }

        // ---- store D: lane holds col n (tile0) / 16+n (tile1), rows 8h+j ----
        const int rbase = rowBase + grow + 8 * h;
        if (fullGroup) {
            float* op = out + rbase * TAPS + n;    // fits 32-bit (B*32 <= INT_MAX)
            #pragma unroll
            for (int j = 0; j < 8; ++j) {
                op[j * TAPS]      = acc0[j];
                op[j * TAPS + 16] = acc1[j];
            }
        } else {
            #pragma unroll
            for (int j = 0; j < 8; ++j) {
                const int rr = rbase + j;
                if (rr < Btotal) {
                    out[rr * TAPS + n]      = acc0[j];
                    out[rr * TAPS + 16 + n] = acc1[j];
                }
            }
        }
        __syncthreads();   // before next group reuses As/Xs
    }
}

extern "C" void kernel_launch(void* const* d_in, const int* in_sizes, int n_in,
                              void* d_out, int out_size, void* d_ws, size_t ws_size,
                              hipStream_t stream) {
    const float* x     = (const float*)d_in[0];   // input_now  [B,1]
    const float* prev  = (const float*)d_in[1];   // output_prev[B,32]
    const float* relax = (const float*)d_in[2];   // relax      [1]
    float* out = (float*)d_out;                   // [B,32]

    const int Btotal = in_sizes[0];
    const int totalGroups = (Btotal + ROWS_PER_GROUP - 1) / ROWS_PER_GROUP;
    const int numBlocks = totalGroups < 1024 ? totalGroups : 1024;
    const int gpb = (totalGroups + numBlocks - 1) / numBlocks;

    laguerre_wmma_kernel<<<numBlocks, 256, 0, stream>>>(
        x, prev, relax, out, Btotal, gpb, numBlocks);
}